// MsCEGroupConv1d_37744172597299
// MI455X (gfx1250) — compile-verified
//
#include <hip/hip_runtime.h>

// MsCE temporal shift-conv + 1x1 channel mix, fused, fp32 WMMA (gfx1250).
//
// x:     [NT=128, C=256, H=28, W=28] f32
// w3:    [128, 3] f32   (channels 0..127,   pad 1)
// w5:    [ 64, 5] f32   (channels 128..191, pad 2)
// w7:    [ 64, 7] f32   (channels 192..255, pad 3)
// net_w: [O=256, C=256] f32
// net_b: [O=256] f32
// out:   [NT, O, H, W] f32

typedef __attribute__((ext_vector_type(2))) float v2f;
typedef __attribute__((ext_vector_type(4))) float v4f;
typedef __attribute__((ext_vector_type(8))) float v8f;

#define NSEG   8
#define C_DIM  256
#define O_DIM  256
#define HW     784      // 28*28 = 49*16
#define NPIX   112      // pixels per workgroup (7 x 16)
#define NTILE  7        // 16-pixel WMMA N-tiles per workgroup
#define KSTEPS (C_DIM / 4)

__global__ __launch_bounds__(256)
void msce_fused_wmma_kernel(const float* __restrict__ x,
                            const float* __restrict__ w3,
                            const float* __restrict__ w5,
                            const float* __restrict__ w7,
                            const float* __restrict__ netw,
                            const float* __restrict__ netb,
                            float* __restrict__ out)
{
    extern __shared__ float ldsY[];          // [C_DIM][NPIX] = 112 KB

    const int block = blockIdx.x;
    const int frame = block / NTILE;         // 0..127  (global time index nt)
    const int pix0  = (block % NTILE) * NPIX;
    const int nb    = frame / NSEG;          // segment (batch) index
    const int t     = frame % NSEG;          // position within segment
    const int tid   = threadIdx.x;           // 0..255

    // ---------------- Phase 1: temporal depthwise conv -> LDS ----------------
    {
        const int c = tid;                   // one channel per thread
        int k, p;
        const float* wrow;
        if (c < C_DIM / 2)                  { k = 3; p = 1; wrow = w3 + c * 3; }
        else if (c < C_DIM / 2 + C_DIM / 4) { k = 5; p = 2; wrow = w5 + (c - C_DIM / 2) * 5; }
        else                                { k = 7; p = 3; wrow = w7 + (c - C_DIM / 2 - C_DIM / 4) * 7; }

        float wv[7];
#pragma unroll
        for (int i = 0; i < 7; ++i) wv[i] = 0.0f;
        for (int i = 0; i < k; ++i) wv[i] = wrow[i];

        const size_t cframe = (size_t)c * HW + (size_t)pix0;

#pragma unroll
        for (int ch = 0; ch < NTILE; ++ch) {
            v4f a0 = {0.f,0.f,0.f,0.f}, a1 = a0, a2 = a0, a3 = a0;
#pragma unroll
            for (int i = 0; i < 7; ++i) {
                const float wgt  = wv[i];
                const int   tsrc = t + i - p;
                if (i < k && wgt != 0.0f && tsrc >= 0 && tsrc < NSEG) {
                    const float* src = x + (size_t)(nb * NSEG + tsrc) * (C_DIM * HW)
                                         + cframe + ch * 16;
                    v4f x0 = *(const v4f*)(src + 0);
                    v4f x1 = *(const v4f*)(src + 4);
                    v4f x2 = *(const v4f*)(src + 8);
                    v4f x3 = *(const v4f*)(src + 12);
                    a0 += wgt * x0; a1 += wgt * x1; a2 += wgt * x2; a3 += wgt * x3;
                }
            }
            float* dst = ldsY + c * NPIX + ch * 16;
            *(v4f*)(dst + 0)  = a0;
            *(v4f*)(dst + 4)  = a1;
            *(v4f*)(dst + 8)  = a2;
            *(v4f*)(dst + 12) = a3;
        }
    }
    __syncthreads();

    // ---------------- Phase 2: out = net_w @ y + b via V_WMMA_F32_16X16X4_F32
    const int wave   = tid >> 5;             // 0..7
    const int lane   = tid & 31;
    const int laneHi = lane >> 4;            // 0: lanes 0-15, 1: lanes 16-31
    const int col    = lane & 15;            // M (for A) / N (for B,C,D) index
    const int o0     = wave * 32;            // this wave: rows [o0, o0+32)

    // C/D layout: VGPR r holds M = r (lanes 0-15) / M = r+8 (lanes 16-31), N = col.
    v8f acc[2][NTILE];
#pragma unroll
    for (int m = 0; m < 2; ++m) {
        v8f binit;
#pragma unroll
        for (int r = 0; r < 8; ++r)
            binit[r] = netb[o0 + m * 16 + r + laneHi * 8];
#pragma unroll
        for (int j = 0; j < NTILE; ++j)
            acc[m][j] = binit;
    }

    // A (16x4 f32) layout: lane M = col; VGPR0/1 = K = (2*laneHi) / (2*laneHi+1).
    // B (4x16 f32) layout: lane N = col; VGPR0/1 = K = (2*laneHi) / (2*laneHi+1).
    const float* aRow0 = netw + (size_t)(o0 + col) * C_DIM;
    const float* aRow1 = netw + (size_t)(o0 + 16 + col) * C_DIM;

#pragma unroll 4
    for (int kk = 0; kk < KSTEPS; ++kk) {
        const int cbh = kk * 4 + laneHi * 2;      // this lane's K-pair base channel
        v2f a0 = *(const v2f*)(aRow0 + cbh);
        v2f a1 = *(const v2f*)(aRow1 + cbh);
#pragma unroll
        for (int j = 0; j < NTILE; ++j) {
            v2f b;
            b.x = ldsY[(cbh + 0) * NPIX + j * 16 + col];
            b.y = ldsY[(cbh + 1) * NPIX + j * 16 + col];
            acc[0][j] = __builtin_amdgcn_wmma_f32_16x16x4_f32(
                false, a0, false, b, (short)0, acc[0][j], false, false);
            acc[1][j] = __builtin_amdgcn_wmma_f32_16x16x4_f32(
                false, a1, false, b, (short)0, acc[1][j], false, false);
        }
    }

    // ---------------- Store D tiles --------------------------------------
    float* outF = out + (size_t)frame * (O_DIM * HW);
#pragma unroll
    for (int m = 0; m < 2; ++m) {
#pragma unroll
        for (int j = 0; j < NTILE; ++j) {
#pragma unroll
            for (int r = 0; r < 8; ++r) {
                const int orow = o0 + m * 16 + r + laneHi * 8;
                outF[(size_t)orow * HW + pix0 + j * 16 + col] = acc[m][j][r];
            }
        }
    }
}

extern "C" void kernel_launch(void* const* d_in, const int* in_sizes, int n_in,
                              void* d_out, int out_size, void* d_ws, size_t ws_size,
                              hipStream_t stream) {
    const float* x    = (const float*)d_in[0];
    const float* w3   = (const float*)d_in[1];
    const float* w5   = (const float*)d_in[2];
    const float* w7   = (const float*)d_in[3];
    const float* netw = (const float*)d_in[4];
    const float* netb = (const float*)d_in[5];
    float* out = (float*)d_out;

    const int grid = 128 * NTILE;                    // 128 frames x 7 pixel-blocks
    const size_t smem = (size_t)C_DIM * NPIX * sizeof(float);  // 112 KB dynamic LDS
    msce_fused_wmma_kernel<<<grid, 256, smem, stream>>>(x, w3, w5, w7, netw, netb, out);
}